// NRCGI_4715874091077
// MI455X (gfx1250) — compile-verified
//
#include <hip/hip_runtime.h>
#include <math.h>

// ---------------------------------------------------------------------------
// DIN-style CTR model for MI455X (gfx1250, wave32).
//  K0 : collapse cross-att weights (vl = Wl@wt[0:16], vr = Wr@wt[16:32]), zero stats
//  Kp : zero-pad w1 (120x200) -> w1p (120x208) so GEMM1 B-loads need no predication
//  K1 : per-batch-row gather + cluster scatter (LDS ds_add_f32) + att + concat(120)
//  K2 : GEMM1 via V_WMMA_F32_16X16X4_F32, fully unrolled K, fused bias + Dice stats
//  K3 : Dice apply
//  K4 : GEMM2 (K=200, N=80) same WMMA path
//  K5 : Dice apply
//  K6 : final 80-dot + sigmoid -> d_out (2048 f32)
// ---------------------------------------------------------------------------

#define SEQLEN  100
#define NCOLS   1202      // 2 + 2*100 + 2*100*5
#define NCLUS   20
#define EMB     8
#define BATCHN  2048
#define CONCATD 120       // EMB*15
#define H1DIM   200
#define H1PAD   208       // H1DIM rounded up to 16
#define H2DIM   80

typedef __attribute__((ext_vector_type(2))) float v2f;
typedef __attribute__((ext_vector_type(8))) float v8f;

// ------------------------- K0: setup ---------------------------------------
// vlvr layout: [m][0:8]=vl, [m][8:16]=vr, m in {u_l1,u_l2,i_l1,i_l2}
__global__ void k0_setup(const float* __restrict__ wl0, const float* __restrict__ wr0, const float* __restrict__ wt0,
                         const float* __restrict__ wl1, const float* __restrict__ wr1, const float* __restrict__ wt1,
                         const float* __restrict__ wl2, const float* __restrict__ wr2, const float* __restrict__ wt2,
                         const float* __restrict__ wl3, const float* __restrict__ wr3, const float* __restrict__ wt3,
                         float* __restrict__ vlvr, float* __restrict__ stats) {
  int tid = threadIdx.x;
  for (int i = tid; i < 560; i += blockDim.x) stats[i] = 0.0f;  // Dice accumulators
  if (tid < 64) {
    int m = tid >> 4, r = tid & 15, which = r >> 3, k = r & 7;
    const float* W; const float* T;
    switch (m) {
      case 0:  W = which ? wr0 : wl0; T = wt0; break;
      case 1:  W = which ? wr1 : wl1; T = wt1; break;
      case 2:  W = which ? wr2 : wl2; T = wt2; break;
      default: W = which ? wr3 : wl3; T = wt3; break;
    }
    float acc = 0.0f;
    #pragma unroll
    for (int j = 0; j < 16; ++j) acc += W[k * 16 + j] * T[which * 16 + j];
    vlvr[m * 16 + which * 8 + k] = acc;
  }
}

// ------------------------- Kp: pad w1 columns 200 -> 208 --------------------
__global__ void k_pad_w1(const float* __restrict__ w1, float* __restrict__ w1p) {
  int i = blockIdx.x * blockDim.x + threadIdx.x;
  if (i >= CONCATD * H1PAD) return;
  int r = i / H1PAD, cidx = i - r * H1PAD;
  w1p[i] = (cidx < H1DIM) ? w1[r * H1DIM + cidx] : 0.0f;
}

// ------------------------- K1: gather + cluster sums + att ------------------
__global__ void __launch_bounds__(128)
k1_features(const int* __restrict__ x, const int* __restrict__ cate,
            const int* __restrict__ ucl, const int* __restrict__ icl,
            const float* __restrict__ ue, const float* __restrict__ ie,
            const float* __restrict__ ce, const float* __restrict__ vlvr,
            const float* __restrict__ bt0, const float* __restrict__ bt1,
            const float* __restrict__ bt2, const float* __restrict__ bt3,
            float* __restrict__ concat_out) {
  __shared__ float cs[4 * NCLUS * EMB];   // cluster sums: [u_l1,u_l2,i_l1,i_l2]
  __shared__ float rep[4 * EMB];
  __shared__ float uem[EMB], tem[EMB];
  const int b = blockIdx.x;
  const int tid = threadIdx.x;
  const int* xr = x + (size_t)b * NCOLS;

  for (int i = tid; i < 4 * NCLUS * EMB; i += blockDim.x) cs[i] = 0.0f;
  if (tid < 4 * EMB) rep[tid] = 0.0f;
  if (tid < EMB) {
    int u = xr[0];
    uem[tid] = ue[u * EMB + tid];
    int tg = xr[601];                                    // target item
    tem[tid] = ie[tg * EMB + tid] + ce[cate[tg] * EMB + tid];
  }
  __syncthreads();

  // 1200 gather positions: [0,100)=u_l1, [100,600)=u_l2, [600,700)=i_l1, [700,1200)=i_l2
  for (int p = tid; p < 1200; p += blockDim.x) {
    int seg, col;
    if (p < 100)      { seg = 0; col = 1 + p; }
    else if (p < 600) { seg = 1; col = 101 + (p - 100); }
    else if (p < 700) { seg = 2; col = 602 + (p - 600); }
    else              { seg = 3; col = 702 + (p - 700); }
    int a = xr[col];
    float e[EMB]; int cl;
    if (seg == 0 || seg == 3) {        // item+cate embedding, item clusters
      const float4* ip = (const float4*)(ie + (size_t)a * EMB);
      const float4* cp = (const float4*)(ce + (size_t)cate[a] * EMB);
      float4 i0 = ip[0], i1 = ip[1], c0 = cp[0], c1 = cp[1];
      e[0]=i0.x+c0.x; e[1]=i0.y+c0.y; e[2]=i0.z+c0.z; e[3]=i0.w+c0.w;
      e[4]=i1.x+c1.x; e[5]=i1.y+c1.y; e[6]=i1.z+c1.z; e[7]=i1.w+c1.w;
      cl = icl[a] - 1;
    } else {                           // user embedding, user clusters
      const float4* up = (const float4*)(ue + (size_t)a * EMB);
      float4 u0 = up[0], u1 = up[1];
      e[0]=u0.x; e[1]=u0.y; e[2]=u0.z; e[3]=u0.w;
      e[4]=u1.x; e[5]=u1.y; e[6]=u1.z; e[7]=u1.w;
      cl = ucl[a] - 1;
    }
    float* dst = &cs[(seg * NCLUS + cl) * EMB];
    #pragma unroll
    for (int k = 0; k < EMB; ++k) atomicAdd(&dst[k], e[k]);   // ds_add_f32
  }
  __syncthreads();

  // 80 (head, cluster) pairs: collapsed cross-attention + weighted rep
  if (tid < 4 * NCLUS) {
    int m = tid / NCLUS;
    const float* s = &cs[tid * EMB];
    const float* t = (m == 0 || m == 3) ? tem : uem;   // u_l1,i_l2 vs item; u_l2,i_l1 vs user
    float att = (m == 0) ? bt0[0] : (m == 1) ? bt1[0] : (m == 2) ? bt2[0] : bt3[0];
    #pragma unroll
    for (int k = 0; k < EMB; ++k) {
      float sk = s[k], tk = t[k];
      att += fabsf(sk - tk) * vlvr[m * 16 + k] + sk * tk * vlvr[m * 16 + 8 + k];
    }
    float aa = tanhf(att);
    #pragma unroll
    for (int k = 0; k < EMB; ++k) atomicAdd(&rep[m * EMB + k], s[k] * aa);
  }
  __syncthreads();

  // concat: [ue, ie, r_ul1, r_il1, r_ul2, r_il2, ue*ie, ue*r_il1, ue*r_il2,
  //          r_ul1*ie, r_ul1*r_il1, r_ul1*r_il2, r_ul2*ie, r_ul2*r_il1, r_ul2*r_il2]
  if (tid < CONCATD) {
    int j = tid / EMB, k = tid % EMB;
    float uek = uem[k], tek = tem[k];
    float r_ul1 = rep[0 * EMB + k], r_ul2 = rep[1 * EMB + k];
    float r_il1 = rep[2 * EMB + k], r_il2 = rep[3 * EMB + k];
    float v;
    switch (j) {
      case 0:  v = uek; break;
      case 1:  v = tek; break;
      case 2:  v = r_ul1; break;
      case 3:  v = r_il1; break;
      case 4:  v = r_ul2; break;
      case 5:  v = r_il2; break;
      case 6:  v = uek * tek; break;
      case 7:  v = uek * r_il1; break;
      case 8:  v = uek * r_il2; break;
      case 9:  v = r_ul1 * tek; break;
      case 10: v = r_ul1 * r_il1; break;
      case 11: v = r_ul1 * r_il2; break;
      case 12: v = r_ul2 * tek; break;
      case 13: v = r_ul2 * r_il1; break;
      default: v = r_ul2 * r_il2; break;
    }
    concat_out[(size_t)b * CONCATD + tid] = v;
  }
}

// ------------------------- K2/K4: WMMA f32 GEMM -----------------------------
// One wave computes one 16x16 tile of H = A(MxK) @ Wp(KxNP) + bias.
// Wp is column-padded (NP multiple of 16, pad = 0) so ALL loads are in-bounds:
// no predication, no exec divergence in the unrolled WMMA chain.
// Compile-time K/NP folds every address into an immediate instruction offset.
// A frag (16x4 f32): lane<16 -> v0=K+0,v1=K+1 ; lane>=16 -> v0=K+2,v1=K+3 ; row=lane&15
// B frag (4x16 f32): mirrored K split, col = n0 + (lane&15)
// C/D (16x16 f32):   row = m0 + vgpr + 8*(lane>=16), col = n0 + (lane&15)
template <int K, int N, int NP>
__global__ void __launch_bounds__(32)
k_gemm_wmma(const float* __restrict__ A, const float* __restrict__ Wp,
            const float* __restrict__ bias, float* __restrict__ H,
            float* __restrict__ colsum, float* __restrict__ colsumsq) {
  const int lane = threadIdx.x;
  const int half = lane >> 4;
  const int ln   = lane & 15;
  const int m0 = blockIdx.x * 16;
  const int n0 = blockIdx.y * 16;
  const int n  = n0 + ln;

  const float* Arow = A + (m0 + ln) * K + 2 * half;   // a.x/a.y adjacent -> b64 load
  const float* Wcol = Wp + 2 * half * NP + n;         // padded: never OOB

  v8f c = {};
  #pragma unroll
  for (int k = 0; k < K; k += 4) {
    v2f a, bf;
    a.x  = Arow[k];
    a.y  = Arow[k + 1];
    bf.x = Wcol[k * NP];          // compile-time immediate offsets
    bf.y = Wcol[k * NP + NP];
    c = __builtin_amdgcn_wmma_f32_16x16x4_f32(false, a, false, bf,
                                              (short)0, c, false, false);
  }

  if (n < N) {                    // store mask only (outside the WMMA chain)
    float bn = bias[n];
    float s = 0.0f, s2 = 0.0f;
    #pragma unroll
    for (int vi = 0; vi < 8; ++vi) {
      float h = c[vi] + bn;
      H[(m0 + vi + 8 * half) * N + n] = h;
      s += h; s2 += h * h;
    }
    atomicAdd(&colsum[n], s);     // fused Dice statistics
    atomicAdd(&colsumsq[n], s2);
  }
}

// ------------------------- K3/K5: Dice apply --------------------------------
__global__ void k_dice(float* __restrict__ H, const float* __restrict__ colsum,
                       const float* __restrict__ colsumsq,
                       const float* __restrict__ alpha, int M, int N) {
  int i = blockIdx.x * blockDim.x + threadIdx.x;
  if (i >= M * N) return;
  int nn = i % N;
  float invM = 1.0f / (float)M;
  float mean = colsum[nn] * invM;
  float var  = colsumsq[nn] * invM - mean * mean;
  float h = H[i];
  float z = (h - mean) * rsqrtf(var + 1e-5f);
  float p = 1.0f / (1.0f + expf(-z));
  H[i] = p * h + (1.0f - p) * alpha[nn] * h;
}

// ------------------------- K6: final dot + sigmoid --------------------------
__global__ void k_final(const float* __restrict__ H2, const float* __restrict__ w3,
                        const float* __restrict__ b3, float* __restrict__ out, int M) {
  int i = blockIdx.x * blockDim.x + threadIdx.x;
  if (i >= M) return;
  float acc = b3[0];
  const float* row = H2 + (size_t)i * H2DIM;
  #pragma unroll 8
  for (int j = 0; j < H2DIM; ++j) acc += row[j] * w3[j];
  out[i] = 1.0f / (1.0f + expf(-acc));
}

// ---------------------------------------------------------------------------
extern "C" void kernel_launch(void* const* d_in, const int* in_sizes, int n_in,
                              void* d_out, int out_size, void* d_ws, size_t ws_size,
                              hipStream_t stream) {
  // setup_inputs() insertion order (params flattened in insertion order)
  const int*   x    = (const int*)d_in[0];
  const int*   cate = (const int*)d_in[1];
  const int*   ucl  = (const int*)d_in[2];
  const int*   icl  = (const int*)d_in[3];
  const float* ue   = (const float*)d_in[4];
  const float* ie   = (const float*)d_in[5];
  const float* ce   = (const float*)d_in[6];
  const float* ul1_wl = (const float*)d_in[7];
  const float* ul1_wr = (const float*)d_in[8];
  const float* ul1_wt = (const float*)d_in[9];
  const float* ul1_bt = (const float*)d_in[10];
  const float* il1_wl = (const float*)d_in[11];
  const float* il1_wr = (const float*)d_in[12];
  const float* il1_wt = (const float*)d_in[13];
  const float* il1_bt = (const float*)d_in[14];
  const float* ul2_wl = (const float*)d_in[15];
  const float* ul2_wr = (const float*)d_in[16];
  const float* ul2_wt = (const float*)d_in[17];
  const float* ul2_bt = (const float*)d_in[18];
  const float* il2_wl = (const float*)d_in[19];
  const float* il2_wr = (const float*)d_in[20];
  const float* il2_wt = (const float*)d_in[21];
  const float* il2_bt = (const float*)d_in[22];
  const float* w1 = (const float*)d_in[23];
  const float* b1 = (const float*)d_in[24];
  const float* a1 = (const float*)d_in[25];
  const float* w2 = (const float*)d_in[26];
  const float* b2 = (const float*)d_in[27];
  const float* a2 = (const float*)d_in[28];
  const float* w3 = (const float*)d_in[29];
  const float* b3 = (const float*)d_in[30];

  // workspace layout (floats): ~3.4 MB total
  float* ws     = (float*)d_ws;
  float* vlvr   = ws;                        // 64
  float* stats  = ws + 64;                   // 560 (zeroed every call in K0)
  float* sum1   = stats;                     // 200
  float* sumsq1 = stats + 200;               // 200
  float* sum2   = stats + 400;               // 80
  float* sumsq2 = stats + 480;               // 80
  float* concat = ws + 640;                  // 2048*120
  float* h1     = concat + BATCHN * CONCATD; // 2048*200
  float* h2     = h1 + BATCHN * H1DIM;       // 2048*80
  float* w1p    = h2 + BATCHN * H2DIM;       // 120*208 zero-padded copy of w1

  // head order m = {u_l1, u_l2, i_l1, i_l2}
  k0_setup<<<1, 128, 0, stream>>>(ul1_wl, ul1_wr, ul1_wt,
                                  ul2_wl, ul2_wr, ul2_wt,
                                  il1_wl, il1_wr, il1_wt,
                                  il2_wl, il2_wr, il2_wt, vlvr, stats);

  k_pad_w1<<<(CONCATD * H1PAD + 255) / 256, 256, 0, stream>>>(w1, w1p);

  k1_features<<<BATCHN, 128, 0, stream>>>(x, cate, ucl, icl, ue, ie, ce, vlvr,
                                          ul1_bt, ul2_bt, il1_bt, il2_bt, concat);

  dim3 g1(BATCHN / 16, H1PAD / 16);          // 128 x 13
  k_gemm_wmma<CONCATD, H1DIM, H1PAD><<<g1, 32, 0, stream>>>(concat, w1p, b1, h1,
                                                            sum1, sumsq1);
  k_dice<<<(BATCHN * H1DIM + 255) / 256, 256, 0, stream>>>(h1, sum1, sumsq1, a1,
                                                           BATCHN, H1DIM);
  dim3 g2(BATCHN / 16, H2DIM / 16);          // 128 x 5
  k_gemm_wmma<H1DIM, H2DIM, H2DIM><<<g2, 32, 0, stream>>>(h1, w2, b2, h2,
                                                          sum2, sumsq2);
  k_dice<<<(BATCHN * H2DIM + 255) / 256, 256, 0, stream>>>(h2, sum2, sumsq2, a2,
                                                           BATCHN, H2DIM);
  k_final<<<(BATCHN + 255) / 256, 256, 0, stream>>>(h2, w3, b3, (float*)d_out, BATCHN);
}